// PosMultiHeadAttention_77824807403941
// MI455X (gfx1250) — compile-verified
//
#include <hip/hip_runtime.h>
#include <math.h>

#define BB 8
#define SS 2048
#define DD 512

typedef float v2f __attribute__((ext_vector_type(2)));
typedef float v8f __attribute__((ext_vector_type(8)));

// Fast tanh: gfx1250 has a hardware V_TANH_F32 trans op.
__device__ __forceinline__ float tanh_fast(float x) {
#if __has_builtin(__builtin_amdgcn_tanhf)
  return __builtin_amdgcn_tanhf(x);
#elif __has_builtin(__builtin_amdgcn_tanh_f32)
  return __builtin_amdgcn_tanh_f32(x);
#else
  // branchless fallback: tanh(x) = 1 - 2/(exp(2x)+1)
  float t = __expf(2.0f * x);
  return 1.0f - 2.0f * __builtin_amdgcn_rcpf(t + 1.0f);
#endif
}

// ---------------------------------------------------------------------------
// K1: wksum[d] = sum_o Wk[o,d];  scal[0] = sum(bk);  scal[1] = bq . bl
// ---------------------------------------------------------------------------
__global__ __launch_bounds__(256) void prep1_kernel(
    const float* __restrict__ Wk, const float* __restrict__ bk,
    const float* __restrict__ bq, const float* __restrict__ bl,
    float* __restrict__ wksum, float* __restrict__ scal) {
  int gid = blockIdx.x * blockDim.x + threadIdx.x;
  if (gid < DD) {
    float s = 0.0f;
    for (int o = 0; o < DD; ++o) s += Wk[o * DD + gid];
    wksum[gid] = s;
  } else if (gid == DD) {
    float s = 0.0f;
    for (int o = 0; o < DD; ++o) s += bk[o];
    scal[0] = s;
  } else if (gid == DD + 1) {
    float s = 0.0f;
    for (int o = 0; o < DD; ++o) s += bq[o] * bl[o];
    scal[1] = s;
  }
}

// ---------------------------------------------------------------------------
// K2: mw[b,o] = sum_j Wl[o,j] * M_pos[b,j]   (one wave per (b,o))
// ---------------------------------------------------------------------------
__global__ __launch_bounds__(32) void mw_kernel(
    const float* __restrict__ Wl, const float* __restrict__ mpos,
    float* __restrict__ mw) {
  int o = blockIdx.x, b = blockIdx.y, lane = threadIdx.x;
  const float* wr = Wl + (size_t)o * SS;
  const float* mp = mpos + (size_t)b * SS;
  float acc = 0.0f;
  for (int j = lane; j < SS; j += 32) acc += wr[j] * mp[j];
  for (int off = 16; off; off >>= 1) acc += __shfl_xor(acc, off, 32);
  if (lane == 0) mw[b * DD + o] = acc;
}

// ---------------------------------------------------------------------------
// K3: v[b,d] = sum_o Wq[o,d]*mw[b,o];  u[d] = sum_o Wq[o,d]*bl[o]
// ---------------------------------------------------------------------------
__global__ __launch_bounds__(256) void v_u_kernel(
    const float* __restrict__ Wq, const float* __restrict__ bl,
    const float* __restrict__ mw, float* __restrict__ v, float* __restrict__ u) {
  int b = blockIdx.y;
  int d = blockIdx.x * 256 + threadIdx.x;
  float acc = 0.0f;
  for (int o = 0; o < DD; ++o) acc += Wq[o * DD + d] * mw[b * DD + o];
  v[b * DD + d] = acc;
  if (b == 0) {
    float au = 0.0f;
    for (int o = 0; o < DD; ++o) au += Wq[o * DD + d] * bl[o];
    u[d] = au;
  }
}

// ---------------------------------------------------------------------------
// K3b: c[b] = bq . mw[b,:]
// ---------------------------------------------------------------------------
__global__ __launch_bounds__(32) void c_kernel(
    const float* __restrict__ bq, const float* __restrict__ mw,
    float* __restrict__ c) {
  int b = blockIdx.x, lane = threadIdx.x;
  float acc = 0.0f;
  for (int o = lane; o < DD; o += 32) acc += bq[o] * mw[b * DD + o];
  for (int off = 16; off; off >>= 1) acc += __shfl_xor(acc, off, 32);
  if (lane == 0) c[b] = acc;
}

// ---------------------------------------------------------------------------
// K4: per (b,i):  qm = query.v[b] + c[b];  qb = query.u + cb;
//                 ksum = key.wksum + bks        (one wave per row)
// ---------------------------------------------------------------------------
__global__ __launch_bounds__(128) void row_dots_kernel(
    const float* __restrict__ query, const float* __restrict__ keyin,
    const float* __restrict__ v, const float* __restrict__ u,
    const float* __restrict__ wksum, const float* __restrict__ c,
    const float* __restrict__ scal, float* __restrict__ qm,
    float* __restrict__ qb, float* __restrict__ ksum) {
  int b = blockIdx.y;
  int i = blockIdx.x * 4 + (threadIdx.x >> 5);
  int lane = threadIdx.x & 31;
  size_t base = ((size_t)b * SS + i) * DD;
  float aqm = 0.0f, aqb = 0.0f, aks = 0.0f;
  for (int d = lane; d < DD; d += 32) {
    float qv = query[base + d];
    float kv = keyin[base + d];
    aqm += qv * v[b * DD + d];
    aqb += qv * u[d];
    aks += kv * wksum[d];
  }
  for (int off = 16; off; off >>= 1) {
    aqm += __shfl_xor(aqm, off, 32);
    aqb += __shfl_xor(aqb, off, 32);
    aks += __shfl_xor(aks, off, 32);
  }
  if (lane == 0) {
    qm[b * SS + i] = aqm + c[b];
    qb[b * SS + i] = aqb + scal[1];
    ksum[b * SS + i] = aks + scal[0];
  }
}

// ---------------------------------------------------------------------------
// K5: scores[b,i,j] = tanh(ksum[b,j]*qm[b,i] + qb[b,i] + bias), row-masked
//     softmax over j.  Rank-1 tile via V_WMMA_F32_16X16X4_F32.
//     Softmax shift = 0 is always safe for unmasked rows (tanh in (-1,1),
//     exp(s) in (1/e, e)); masked rows are uniform 1/S.  Branchless:
//     pass 1 sums exp(tanh), pass 2 recomputes + normalizes + stores.
// ---------------------------------------------------------------------------
__global__ __launch_bounds__(256) void scores_softmax_kernel(
    const float* __restrict__ qm, const float* __restrict__ qb,
    const float* __restrict__ ksum, const float* __restrict__ mpos,
    const float* __restrict__ attn_bias, float* __restrict__ out) {
  __shared__ float qm_s[16], qb_s[16], rmask_s[16];
  __shared__ float ws_s[8][16];
  __shared__ float rowinv_s[16];

  const int b = blockIdx.y;
  const int i0 = blockIdx.x * 16;
  const int tid = threadIdx.x;
  const int lane = tid & 31;
  const int w = tid >> 5;
  const int half = (lane >> 4) & 1;
  const int col = lane & 15;
  const float bias = attn_bias[0];

  if (tid < 16) {
    qm_s[tid] = qm[b * SS + i0 + tid];
    qb_s[tid] = qb[b * SS + i0 + tid];
    rmask_s[tid] = mpos[b * SS + i0 + tid];
  }
  __syncthreads();

  // Loop-invariant A operand: 16x4 f32, only K=0 column nonzero (qm rows).
  v2f a;
  a.x = (half == 0) ? qm_s[col] : 0.0f;
  a.y = 0.0f;

  float qbr[8];
#pragma unroll
  for (int r = 0; r < 8; ++r) qbr[r] = qb_s[r + half * 8] + bias;

  v8f czero = {};

  // ---- Pass 1: rowsum of exp(tanh(score)) --------------------------------
  float rsum[8];
#pragma unroll
  for (int r = 0; r < 8; ++r) rsum[r] = 0.0f;

  for (int jt = w; jt < SS / 16; jt += 8) {
    int j0 = jt * 16;
    float kv = ksum[b * SS + j0 + col];
    v2f bm;
    bm.x = (half == 0) ? kv : 0.0f;
    bm.y = 0.0f;
    v8f d = __builtin_amdgcn_wmma_f32_16x16x4_f32(
        false, a, false, bm, (short)0, czero, false, false);
#pragma unroll
    for (int r = 0; r < 8; ++r) {
      rsum[r] += __expf(tanh_fast(d[r] + qbr[r]));
    }
  }

  // reduce across the 16 lanes of each half-wave (rows are per-half)
#pragma unroll
  for (int off = 1; off < 16; off <<= 1) {
#pragma unroll
    for (int r = 0; r < 8; ++r) rsum[r] += __shfl_xor(rsum[r], off, 32);
  }
  if (col == 0) {
#pragma unroll
    for (int r = 0; r < 8; ++r) ws_s[w][r + half * 8] = rsum[r];
  }
  __syncthreads();

  // combine the 8 wave partials per row
  if (tid < 16) {
    float s = 0.0f;
    for (int ww = 0; ww < 8; ++ww) s += ws_s[ww][tid];
    rowinv_s[tid] = 1.0f / s;
  }
  __syncthreads();

  float rin[8];
  bool mk[8];
#pragma unroll
  for (int r = 0; r < 8; ++r) {
    int row = r + half * 8;
    rin[r] = rowinv_s[row];
    mk[r] = (rmask_s[row] == 0.0f);  // masked row -> uniform 1/S
  }

  // ---- Pass 2: recompute tiles (WMMA), normalize, store ------------------
  const float unif = 1.0f / (float)SS;
  for (int jt = w; jt < SS / 16; jt += 8) {
    int j0 = jt * 16;
    float kv = ksum[b * SS + j0 + col];
    v2f bm;
    bm.x = (half == 0) ? kv : 0.0f;
    bm.y = 0.0f;
    v8f d = __builtin_amdgcn_wmma_f32_16x16x4_f32(
        false, a, false, bm, (short)0, czero, false, false);
#pragma unroll
    for (int r = 0; r < 8; ++r) {
      float e = __expf(tanh_fast(d[r] + qbr[r])) * rin[r];
      float val = mk[r] ? unif : e;
      int row = r + half * 8;
      out[(size_t)b * SS * SS + (size_t)(i0 + row) * SS + j0 + col] = val;
    }
  }
}

// ---------------------------------------------------------------------------
extern "C" void kernel_launch(void* const* d_in, const int* in_sizes, int n_in,
                              void* d_out, int out_size, void* d_ws,
                              size_t ws_size, hipStream_t stream) {
  (void)in_sizes; (void)n_in; (void)out_size; (void)ws_size;
  const float* query = (const float*)d_in[0];
  const float* keyin = (const float*)d_in[1];
  const float* mpos  = (const float*)d_in[2];
  const float* Wq    = (const float*)d_in[3];
  const float* bq    = (const float*)d_in[4];
  const float* Wk    = (const float*)d_in[5];
  const float* bk    = (const float*)d_in[6];
  const float* abias = (const float*)d_in[7];
  const float* Wl    = (const float*)d_in[8];
  const float* bl    = (const float*)d_in[9];
  float* out = (float*)d_out;

  float* ws    = (float*)d_ws;
  float* wksum = ws;            // 512
  float* scal  = ws + 512;      // 2  : [sum(bk), bq.bl]
  float* c     = ws + 516;      // 8
  float* u     = ws + 1024;     // 512
  float* mw    = ws + 1536;     // 8*512
  float* v     = ws + 5632;     // 8*512
  float* qm    = ws + 9728;     // 8*2048
  float* qb    = ws + 26112;    // 8*2048
  float* ks    = ws + 42496;    // 8*2048

  prep1_kernel<<<3, 256, 0, stream>>>(Wk, bk, bq, bl, wksum, scal);
  mw_kernel<<<dim3(512, 8), 32, 0, stream>>>(Wl, mpos, mw);
  v_u_kernel<<<dim3(2, 8), 256, 0, stream>>>(Wq, bl, mw, v, u);
  c_kernel<<<8, 32, 0, stream>>>(bq, mw, c);
  row_dots_kernel<<<dim3(512, 8), 128, 0, stream>>>(query, keyin, v, u, wksum,
                                                    c, scal, qm, qb, ks);
  scores_softmax_kernel<<<dim3(128, 8), 256, 0, stream>>>(qm, qb, ks, mpos,
                                                          abias, out);
}